// HungarianMatcher2_13649406066908
// MI455X (gfx1250) — compile-verified
//
#include <hip/hip_runtime.h>
#include <hip/hip_bf16.h>

typedef __attribute__((ext_vector_type(2))) float v2f;
typedef __attribute__((ext_vector_type(8))) float v8f;

namespace {
constexpr int kBS = 8;
constexpr int kNQ = 1024;   // queries (columns of solver matrix)
constexpr int kT  = 256;    // targets (rows of solver matrix)
constexpr float kCostPoint = 0.05f;
}

// ---------------------------------------------------------------------------
// Kernel 1: cost matrix C[b][i(target)][j(query)] via WMMA f32 16x16x4.
// dist^2 = a.a + b.b - 2 a.b computed exactly as a rank-4 product:
//   A row  (target i): ( t0, t1, t0^2+t1^2, 1 )
//   B col  (query  j): ( -2q0, -2q1, 1, q0^2+q1^2 )
// One V_WMMA_F32_16X16X4_F32 per 16x16 tile per wave.
// ---------------------------------------------------------------------------
__global__ __launch_bounds__(32) void hm_cost_kernel(
    const float* __restrict__ pred_logits,   // [BS,NQ,2]
    const float* __restrict__ pred_points,   // [BS,NQ,2]
    const float* __restrict__ tgt_points,    // [BS,T,2]
    const int*   __restrict__ tgt_labels,    // [BS,T]
    const int*   __restrict__ img_h_p,
    const int*   __restrict__ img_w_p,
    float*       __restrict__ cost)          // [BS,T,NQ]
{
    const int lane = threadIdx.x;            // 0..31
    const int half = lane >> 4;              // 0: K={0,1}, 1: K={2,3}
    const int lm   = lane & 15;

    int tile = blockIdx.x;
    const int tn = tile & 63;  tile >>= 6;   // 64 tiles over NQ
    const int tm = tile & 15;  tile >>= 4;   // 16 tiles over T
    const int b  = tile;                     // batch

    const float H = (float)img_h_p[0];
    const float W = (float)img_w_p[0];

    // A matrix 16x4 f32: lanes 0-15 hold rows M=lm with K=0,1; lanes 16-31 K=2,3.
    const int   ti = tm * 16 + lm;
    const float t0 = tgt_points[((size_t)b * kT + ti) * 2 + 0];
    const float t1 = tgt_points[((size_t)b * kT + ti) * 2 + 1];
    v2f a;
    a.x = half ? (t0 * t0 + t1 * t1) : t0;
    a.y = half ? 1.0f                : t1;

    // B matrix 4x16 f32: lanes 0-15 hold cols N=lm with K=0,1; lanes 16-31 K=2,3.
    const int   qi = tn * 16 + lm;
    const float q0 = pred_points[((size_t)b * kNQ + qi) * 2 + 0] * H;
    const float q1 = pred_points[((size_t)b * kNQ + qi) * 2 + 1] * W;
    v2f bb;
    bb.x = half ? 1.0f                : (-2.0f * q0);
    bb.y = half ? (q0 * q0 + q1 * q1) : (-2.0f * q1);

    v8f c = {};
    v8f d = __builtin_amdgcn_wmma_f32_16x16x4_f32(
        /*neg_a=*/false, a, /*neg_b=*/false, bb,
        /*c_mod=*/(short)0, c, /*reuse_a=*/false, /*reuse_b=*/false);

    // Class probabilities for this lane's query column (n = lm): 2-way softmax.
    const float l0  = pred_logits[((size_t)b * kNQ + qi) * 2 + 0];
    const float l1  = pred_logits[((size_t)b * kNQ + qi) * 2 + 1];
    const float mx  = fmaxf(l0, l1);
    const float e0  = __expf(l0 - mx);
    const float e1  = __expf(l1 - mx);
    const float inv = 1.0f / (e0 + e1);
    const float p0  = e0 * inv;
    const float p1  = e1 * inv;

    // D layout: lane L, VGPR r -> (m = r + 8*(L>=16), n = L%16)
#pragma unroll
    for (int r = 0; r < 8; ++r) {
        const int   m   = r + half * 8;
        const int   i   = tm * 16 + m;
        const int   lab = tgt_labels[(size_t)b * kT + i];
        const float d2  = fmaxf(d[r], 0.0f);          // guard tiny negatives
        const float val = kCostPoint * sqrtf(d2) - (lab ? p1 : p0);
        cost[((size_t)b * kT + i) * kNQ + tn * 16 + lm] = val;
    }
}

// ---------------------------------------------------------------------------
// Kernel 2: exact JV shortest-augmenting-path LSA, one wave per batch.
// 32 lanes each own columns j = lane + 32*s (s = 0..31). SC/SR kept as
// per-lane register bitmasks (lane-owned -> no cross-lane hazards in the hot
// loop); argmin with first-index tiebreak via shfl_xor butterfly.
// ---------------------------------------------------------------------------
__global__ __launch_bounds__(32) void hm_jv_kernel(
    const float* __restrict__ cost,   // [BS,T,NQ]
    int*         __restrict__ out)    // pred_idx[BS,T] then tgt_idx[BS,T]
{
    const int b    = blockIdx.x;
    const int lane = threadIdx.x;     // 0..31
    const float* __restrict__ C = cost + (size_t)b * kT * kNQ;

    __shared__ float u[kT];
    __shared__ float v[kNQ];
    __shared__ float spc[kNQ];
    __shared__ int   path[kNQ];
    __shared__ int   row4col[kNQ];
    __shared__ int   col4row[kT];

    const float INF = __builtin_huge_valf();

#pragma unroll
    for (int s = 0; s < kNQ / 32; ++s) {
        const int j = lane + (s << 5);
        v[j] = 0.0f;
        row4col[j] = -1;
    }
#pragma unroll
    for (int s = 0; s < kT / 32; ++s) {
        const int k = lane + (s << 5);
        u[k] = 0.0f;
        col4row[k] = -1;
    }
    __syncthreads();

    for (int cur = 0; cur < kT; ++cur) {
        // per-row reset
#pragma unroll
        for (int s = 0; s < kNQ / 32; ++s) {
            const int j = lane + (s << 5);
            spc[j]  = INF;
            path[j] = 0;
        }
        unsigned scMask = 0u;   // bit s: column lane+32*s scanned
        unsigned srMask = 0u;   // bit s: row    lane+32*s visited
        __syncthreads();

        int   i    = cur;
        float minv = 0.0f;
        int   sink = -1;

        while (sink < 0) {
            if (lane == (i & 31)) srMask |= (1u << (i >> 5));   // SR[i] = 1
            const float base = minv - u[i];                      // uniform
            const float* __restrict__ Crow = C + (size_t)i * kNQ;

            float bv = INF;
            int   bj = 0;
#pragma unroll
            for (int s = 0; s < kNQ / 32; ++s) {
                const int  j  = lane + (s << 5);
                const bool sc = (scMask >> s) & 1u;
                float sp = spc[j];
                const float r = base + Crow[j] - v[j];
                if (!sc && r < sp) { sp = r; spc[j] = r; path[j] = i; }
                const float m = sc ? INF : sp;
                if (m < bv) { bv = m; bj = j; }   // j increases with s: first-min kept
            }
            // wave argmin, tiebreak = smallest global column index (jnp.argmin)
#pragma unroll
            for (int off = 16; off > 0; off >>= 1) {
                const float ov = __shfl_xor(bv, off, 32);
                const int   oj = __shfl_xor(bj, off, 32);
                if (ov < bv || (ov == bv && oj < bj)) { bv = ov; bj = oj; }
            }
            minv = bv;                                           // uniform
            if (lane == (bj & 31)) scMask |= (1u << (bj >> 5));  // SC[bj] = 1
            const int rc = row4col[bj];                          // uniform read
            if (rc < 0) sink = bj;
            else        i    = rc;
        }

        __syncthreads();   // spc writes visible for cross-lane gathers below

        // dual update: v[j] -= SC[j] ? (minv - spc[j]) : 0
#pragma unroll
        for (int s = 0; s < kNQ / 32; ++s) {
            if ((scMask >> s) & 1u) {
                const int j = lane + (s << 5);
                v[j] -= (minv - spc[j]);
            }
        }
        // dual update: u[cur] += minv ; u[k] += minv - spc[col4row[k]] for SR rows
#pragma unroll
        for (int s = 0; s < kT / 32; ++s) {
            const int k = lane + (s << 5);
            if (k == cur) {
                u[k] += minv;
            } else if ((srMask >> s) & 1u) {
                int c4 = col4row[k];
                if (c4 < 0) c4 = 0;   // mirrors reference clip (SR rows are assigned)
                u[k] += minv - spc[c4];
            }
        }
        __syncthreads();

        // augment along the alternating path (scalar, lane 0)
        if (lane == 0) {
            int snk = sink;
            while (true) {
                const int pi  = path[snk];
                row4col[snk]  = pi;
                const int nxt = col4row[pi];
                col4row[pi]   = snk;
                snk = nxt;
                if (pi == cur) break;
            }
        }
        __syncthreads();
    }

    // argsort col4row (values are distinct): rank by counting, write both outputs
#pragma unroll
    for (int s = 0; s < kT / 32; ++s) {
        const int k = lane + (s << 5);
        const int c = col4row[k];
        int rank = 0;
        for (int m2 = 0; m2 < kT; ++m2) rank += (col4row[m2] < c) ? 1 : 0;
        out[(size_t)b * kT + rank]             = c;   // pred_idx
        out[(size_t)kBS * kT + b * kT + rank]  = k;   // tgt_idx
    }
}

// ---------------------------------------------------------------------------
extern "C" void kernel_launch(void* const* d_in, const int* in_sizes, int n_in,
                              void* d_out, int out_size, void* d_ws, size_t ws_size,
                              hipStream_t stream) {
    (void)in_sizes; (void)n_in; (void)out_size; (void)ws_size;

    const float* pred_logits = (const float*)d_in[0];   // [8,1024,2]
    const float* pred_points = (const float*)d_in[1];   // [8,1024,2]
    const float* tgt_points  = (const float*)d_in[2];   // [8,256,2]
    const int*   tgt_labels  = (const int*)d_in[3];     // [8,256]
    const int*   img_h       = (const int*)d_in[4];     // scalar
    const int*   img_w       = (const int*)d_in[5];     // scalar

    float* cost = (float*)d_ws;                         // 8*256*1024*4 = 8 MB
    int*   out  = (int*)d_out;                          // 2*8*256 int32

    // 8 batches * 16 M-tiles * 64 N-tiles, one wave per 16x16 WMMA tile
    hm_cost_kernel<<<kBS * (kT / 16) * (kNQ / 16), 32, 0, stream>>>(
        pred_logits, pred_points, tgt_points, tgt_labels, img_h, img_w, cost);

    // one wave-synchronous wave per batch
    hm_jv_kernel<<<kBS, 32, 0, stream>>>(cost, out);
}